// CTCLoss_76158360092759
// MI455X (gfx1250) — compile-verified
//
#include <hip/hip_runtime.h>
#include <stdint.h>

// Problem constants (from reference): T=1000, B=64, C=28, S=200
#define T_STEPS 1000
#define BATCH   64
#define NCLS    28
#define SMAX    200
#define NS      399            // 2*S - 1 states
#define NTHR    128            // 4 wave32 per workgroup
#define L2E     1.44269504088896340736f

typedef unsigned int u32x4 __attribute__((ext_vector_type(4)));
typedef int          i32x8 __attribute__((ext_vector_type(8)));
typedef int          i32x4 __attribute__((ext_vector_type(4)));

// One workgroup per batch element b.
// Phase 1: TDM DMA of prediction[:, b, :] (1000 x 28 f32, row stride B*C) into LDS.
// Phase 2: 1000-step alpha recurrence, double-buffered in LDS, 1 barrier/step.
__global__ __launch_bounds__(NTHR) void ctc_alpha_kernel(
    const float* __restrict__ pred,
    const int*   __restrict__ target,
    const int*   __restrict__ pred_lens,
    const int*   __restrict__ target_lens,
    float*       __restrict__ per_b)
{
    __shared__ float ldsPred[T_STEPS * NCLS];     // 112000 B
    __shared__ float alphaS[2 * (NS + 2)];        // double buffer, 2-elem pad at front of each

    const int b   = blockIdx.x;
    const int tid = threadIdx.x;

    // ---------- Phase 1: Tensor Data Mover bulk copy (wave 0 issues it) ----------
    if (tid < 32) {
        uint64_t gaddr = (uint64_t)(uintptr_t)(pred + (size_t)b * NCLS);
        uint32_t laddr = (uint32_t)(uintptr_t)(&ldsPred[0]);   // low 32 bits = LDS byte offset

        u32x4 g0;
        g0[0] = 1u;                                            // count=1, gather off
        g0[1] = laddr;                                         // lds_addr[63:32]
        g0[2] = (uint32_t)(gaddr & 0xFFFFFFFFu);               // global_addr lo
        g0[3] = (uint32_t)((gaddr >> 32) & 0x1FFFFFFu)         // global_addr[56:32]
              | (2u << 30);                                    // type=2 ("image")

        i32x8 g1;
        g1[0] = (int)(2u << 16);                               // data_size = 4 bytes
        g1[1] = (int)(((uint32_t)NCLS & 0xFFFFu) << 16);       // tensor_dim0 lo16 (=28)
        g1[2] = (int)((((uint32_t)NCLS >> 16) & 0xFFFFu)       // tensor_dim0 hi16
              | (((uint32_t)T_STEPS & 0xFFFFu) << 16));        // tensor_dim1 lo16 (=1000)
        g1[3] = (int)((((uint32_t)T_STEPS >> 16) & 0xFFFFu)    // tensor_dim1 hi16
              | (((uint32_t)NCLS & 0xFFFFu) << 16));           // tile_dim0 = 28
        g1[4] = (int)((uint32_t)T_STEPS & 0xFFFFu);            // tile_dim1 = 1000, tile_dim2 = 0
        g1[5] = (int)((uint32_t)(BATCH * NCLS));               // tensor_dim0_stride lo32 = 1792 elems
        g1[6] = 0;                                             // stride hi16 / dim1_stride lo16
        g1[7] = 0;

        i32x4 gz4 = {0, 0, 0, 0};                              // groups 2/3: unused dims (2D tensor)
        i32x8 gz8 = {0, 0, 0, 0, 0, 0, 0, 0};                  // extra group (clang-23 6-arg form)
        __builtin_amdgcn_tensor_load_to_lds(g0, g1, gz4, gz4, gz8, 0);
        __builtin_amdgcn_s_wait_tensorcnt(0);
    }
    __syncthreads();

    // ---------- Phase 2: recurrence ----------
    // States interleaved: thread tid owns s = tid + 128*j, j = 0..3 (conflict-free LDS).
    int  lab[4];
    bool alive[4];
#pragma unroll
    for (int j = 0; j < 4; ++j) {
        int s    = tid + NTHR * j;
        alive[j] = (s < NS);
        int li   = NCLS - 1;                                   // blank (odd states)
        if (alive[j] && ((s & 1) == 0)) {
            int tg = target[(size_t)b * SMAX + (s >> 1)] - 1;  // tgt = max(target-1, 0)
            li = tg > 0 ? tg : 0;
        }
        lab[j] = li;
    }
    const int idx_t = pred_lens[b] - 1;
    const int s_cap = 2 * target_lens[b] - 2;

    float* bufA = alphaS + 2;
    float* bufB = alphaS + (NS + 2) + 2;
    const float NEG = -__builtin_inff();

    // t = 0: alpha0 = emit[0] (base-2 scaled)
    {
        const float* row = ldsPred;
#pragma unroll
        for (int j = 0; j < 4; ++j) {
            int s = tid + NTHR * j;
            if (alive[j]) bufA[s] = row[lab[j]] * L2E;
        }
    }
    __syncthreads();

    float cap = 0.0f;
    if (tid == 0 && idx_t == 0) cap = bufA[s_cap];

    float* cur = bufA;
    float* nxt = bufB;
    for (int t = 1; t < T_STEPS; ++t) {
        const float* row = ldsPred + t * NCLS;
#pragma unroll
        for (int j = 0; j < 4; ++j) {
            int s = tid + NTHR * j;
            if (!alive[j]) continue;
            float a  = cur[s];
            float a1 = cur[s - 1];                 // pad makes s=0 read legal; masked below
            float a2 = cur[s - 2];
            if (s < 1) a1 = NEG;
            if ((s & 1) || s < 2) a2 = NEG;        // skip allowed only for even s >= 2
            float m   = fmaxf(a, fmaxf(a1, a2));
            float sum = __builtin_amdgcn_exp2f(a  - m)
                      + __builtin_amdgcn_exp2f(a1 - m)
                      + __builtin_amdgcn_exp2f(a2 - m);
            nxt[s] = row[lab[j]] * L2E + m + __builtin_amdgcn_logf(sum);
        }
        __syncthreads();
        float* tmp = cur; cur = nxt; nxt = tmp;    // cur now holds alpha[t]
        if (tid == 0 && t == idx_t) cap = cur[s_cap];
    }

    if (tid == 0) per_b[b] = cap * (1.0f / L2E);   // back to natural log
}

// Deterministic fixed-order reduction of the 64 per-batch values.
__global__ void ctc_reduce_kernel(const float* __restrict__ per_b,
                                  float* __restrict__ out)
{
    if (blockIdx.x == 0 && threadIdx.x == 0) {
        float s = 0.0f;
        for (int i = 0; i < BATCH; ++i) s += per_b[i];
        out[0] = s;
    }
}

extern "C" void kernel_launch(void* const* d_in, const int* in_sizes, int n_in,
                              void* d_out, int out_size, void* d_ws, size_t ws_size,
                              hipStream_t stream) {
    (void)in_sizes; (void)n_in; (void)out_size; (void)ws_size;
    const float* pred   = (const float*)d_in[0];   // (T, B, C) f32 log-softmax
    const int*   target = (const int*)d_in[1];     // (B, S) i32
    const int*   plens  = (const int*)d_in[2];     // (B,)  i32
    const int*   tlens  = (const int*)d_in[3];     // (B,)  i32
    float* per_b = (float*)d_ws;                   // 64 floats scratch

    ctc_alpha_kernel<<<BATCH, NTHR, 0, stream>>>(pred, target, plens, tlens, per_b);
    ctc_reduce_kernel<<<1, 32, 0, stream>>>(per_b, (float*)d_out);
}